// GCNNodeClassifier_24773371364082
// MI455X (gfx1250) — compile-verified
//
#include <hip/hip_runtime.h>

typedef __attribute__((ext_vector_type(2))) float v2f;
typedef __attribute__((ext_vector_type(8))) float v8f;

#define D_IN 512
#define D_H  32
#define N_CLS 40
#define PN_EPS 1e-5f

// ---------------- degree / norm ----------------
__global__ void k_fill1(float* __restrict__ d, int n) {
    int i = blockIdx.x * blockDim.x + threadIdx.x;
    if (i < n) d[i] = 1.0f;                 // self-loop contributes 1 to degree
}

__global__ void k_deg(const int* __restrict__ dst, float* __restrict__ deg, int E) {
    int e = blockIdx.x * blockDim.x + threadIdx.x;
    if (e < E) atomicAdd(&deg[dst[e]], 1.0f);
}

__global__ void k_rsqrt(float* __restrict__ d, int n) {
    int i = blockIdx.x * blockDim.x + threadIdx.x;
    if (i < n) d[i] = rsqrtf(d[i]);         // deg >= 1 always (self-loops)
}

// ---------------- WMMA GEMM: Out[N x 32] = A[N x K] @ W[K x 32] ----------------
// Each wave computes a 16-row x 32-col tile with two V_WMMA_F32_16X16X4_F32
// accumulators. A-fragment (16x4 f32): lanes 0-15 hold K=0,1; lanes 16-31 hold
// K=2,3 for row M=lane%16 -> one aligned 8-byte global load per K-step lands
// directly in the fragment register pair. B-fragments: W is pre-swizzled into
// LDS as float4 Wq[k2][n] = {W[2k2][n], W[2k2+1][n], W[2k2][n+16], W[2k2+1][n+16]}
// so one ds_load_b128 per K-step yields BOTH 16-wide column tiles in two
// even-aligned VGPR pairs (.xy / .zw) with zero repacking.
// C/D layout: VGPR r -> M = r + 8*(lane/16), N = lane%16.
template <int K>
__global__ __launch_bounds__(256) void k_gemm_wmma(const float* __restrict__ A,
                                                   const float* __restrict__ W,
                                                   float* __restrict__ Out, int N) {
    __shared__ float4 Wq[(K / 2) * 16];
    for (int idx = threadIdx.x; idx < (K / 2) * 16; idx += 256) {
        int k2 = idx >> 4;
        int c  = idx & 15;
        float4 q;
        q.x = W[(2 * k2) * D_H + c];
        q.y = W[(2 * k2 + 1) * D_H + c];
        q.z = W[(2 * k2) * D_H + 16 + c];
        q.w = W[(2 * k2 + 1) * D_H + 16 + c];
        Wq[idx] = q;
    }
    __syncthreads();

    const int lane = threadIdx.x & 31;
    const int wave = threadIdx.x >> 5;
    const int m0 = (blockIdx.x * 8 + wave) * 16;
    if (m0 >= N) return;                    // wave-uniform exit

    const int hi  = lane >> 4;              // 0: lanes 0-15, 1: lanes 16-31
    const int klo = hi * 2;
    const int n   = lane & 15;
    const int mr  = m0 + n;
    const int row = mr < N ? mr : N - 1;    // clamp loads on tail (EXEC stays all-1)
    const v2f* ap = (const v2f*)(A + (size_t)row * K + klo);   // 8B-aligned
    const float4* bp = &Wq[hi * 16 + n];    // this lane's B stream (k2 = 2s + hi)

    v8f c0 = {}; v8f c1 = {};
    for (int s = 0; s < K / 4; ++s) {
        v2f a = ap[2 * s];                  // {A[row][4s+klo], A[row][4s+klo+1]}
        float4 bq = bp[32 * s];             // both B fragments, one ds_load_b128
        v2f b0; b0.x = bq.x; b0.y = bq.y;
        v2f b1; b1.x = bq.z; b1.y = bq.w;
        c0 = __builtin_amdgcn_wmma_f32_16x16x4_f32(false, a, false, b0, (short)0, c0, false, false);
        c1 = __builtin_amdgcn_wmma_f32_16x16x4_f32(false, a, false, b1, (short)0, c1, false, false);
    }

    const int mb = m0 + hi * 8;
    if (m0 + 16 <= N) {
        // full tile (all but the last block): unguarded constant-offset stores
        float* outp = Out + (size_t)mb * D_H + n;
#pragma unroll
        for (int r = 0; r < 8; ++r) {
            outp[r * D_H]      = c0[r];
            outp[r * D_H + 16] = c1[r];
        }
    } else {
#pragma unroll
        for (int r = 0; r < 8; ++r) {
            int m = mb + r;
            if (m < N) {
                Out[(size_t)m * D_H + n]      = c0[r];
                Out[(size_t)m * D_H + 16 + n] = c1[r];
            }
        }
    }
}

// ---------------- aggregation ----------------
// agg[i][c] = h[i][c] * dinv[i]^2 + bias[c]   (self-loop message + bias)
template <int C>
__global__ void k_initagg(const float* __restrict__ h, const float* __restrict__ dinv,
                          const float* __restrict__ bias, float* __restrict__ agg, int N) {
    int t = blockIdx.x * blockDim.x + threadIdx.x;
    int i = t / C, c = t % C;
    if (i < N) {
        float dv = dinv[i];
        agg[(size_t)i * C + c] = h[(size_t)i * C + c] * dv * dv + bias[c];
    }
}

// one wave per edge: agg[dst] += h[src] * dinv[src]*dinv[dst]
template <int C>
__global__ void k_scatter(const float* __restrict__ h, const int* __restrict__ src,
                          const int* __restrict__ dst, const float* __restrict__ dinv,
                          float* __restrict__ agg, int E) {
    int t = blockIdx.x * blockDim.x + threadIdx.x;
    int e = t >> 5;
    int lane = t & 31;
    if (e >= E) return;
    // prefetch the streamed edge lists ahead (global_prefetch_b8)
    if (lane == 0 && e + 2048 < E) {
        __builtin_prefetch(&src[e + 2048], 0, 0);
        __builtin_prefetch(&dst[e + 2048], 0, 0);
    }
    int s = src[e], d = dst[e];
    float nrm = dinv[s] * dinv[d];
    for (int c = lane; c < C; c += 32)
        atomicAdd(&agg[(size_t)d * C + c], h[(size_t)s * C + c] * nrm);
}

// ---------------- PairNorm ----------------
// stats[0..31] = column sums, stats[32] = sum of squares (all elements)
__global__ void k_stats(const float* __restrict__ x, float* __restrict__ stats, int N) {
    __shared__ float scol[D_H];
    __shared__ float ssq;
    if (threadIdx.x < D_H) scol[threadIdx.x] = 0.f;
    if (threadIdx.x == 0) ssq = 0.f;
    __syncthreads();
    int c = threadIdx.x & 31;
    int rowStart  = blockIdx.x * (blockDim.x >> 5) + (threadIdx.x >> 5);
    int rowStride = gridDim.x * (blockDim.x >> 5);
    float cs = 0.f, sq = 0.f;
    for (int i = rowStart; i < N; i += rowStride) {
        float v = x[(size_t)i * D_H + c];
        cs += v; sq += v * v;
    }
    atomicAdd(&scol[c], cs);
    atomicAdd(&ssq, sq);
    __syncthreads();
    if (threadIdx.x < D_H) atomicAdd(&stats[threadIdx.x], scol[threadIdx.x]);
    if (threadIdx.x == 0) atomicAdd(&stats[D_H], ssq);
}

// stats -> stats[0..31] = mean, stats[32] = rsqrt(eps + var)
__global__ void k_finalize(float* __restrict__ stats, int N) {
    if (blockIdx.x == 0 && threadIdx.x == 0) {
        float invN = 1.0f / (float)N;
        float msq = 0.f;
        float mean[D_H];
        for (int c = 0; c < D_H; ++c) { mean[c] = stats[c] * invN; msq += mean[c] * mean[c]; }
        float var = stats[D_H] * invN - msq;   // E[sum_c (x-m)^2] = sumsq/N - sum m^2
        float scale = rsqrtf(PN_EPS + var);
        for (int c = 0; c < D_H; ++c) stats[c] = mean[c];
        stats[D_H] = scale;
    }
}

// x = relu((x - mean[c]) * scale), in place
__global__ void k_applypn(float* __restrict__ x, const float* __restrict__ stats, int N) {
    int t = blockIdx.x * blockDim.x + threadIdx.x;
    int i = t >> 5, c = t & 31;
    if (i < N) {
        float v = (x[(size_t)i * D_H + c] - stats[c]) * stats[D_H];
        x[(size_t)i * D_H + c] = v > 0.f ? v : 0.f;
    }
}

// ---------------- final 32 -> 40 GEMM (tiny, memory bound) ----------------
__global__ void k_gemm_final(const float* __restrict__ h, const float* __restrict__ Wf,
                             float* __restrict__ out, int N) {
    __shared__ float Ws[D_H * N_CLS];
    for (int i = threadIdx.x; i < D_H * N_CLS; i += blockDim.x) Ws[i] = Wf[i];
    __syncthreads();
    int t = blockIdx.x * blockDim.x + threadIdx.x;
    int i = t / N_CLS, c = t % N_CLS;
    if (i < N) {
        const float* hr = h + (size_t)i * D_H;
        float acc = 0.f;
#pragma unroll
        for (int k = 0; k < D_H; ++k) acc += hr[k] * Ws[k * N_CLS + c];
        out[(size_t)i * N_CLS + c] = acc;
    }
}

// ---------------- host sequencing ----------------
extern "C" void kernel_launch(void* const* d_in, const int* in_sizes, int n_in,
                              void* d_out, int out_size, void* d_ws, size_t ws_size,
                              hipStream_t stream) {
    const float* x  = (const float*)d_in[0];
    const int* edge = (const int*)d_in[1];   // [2, E]
    const float* W0 = (const float*)d_in[2];
    const float* b0 = (const float*)d_in[3];
    const float* W1 = (const float*)d_in[4];
    const float* b1 = (const float*)d_in[5];
    const float* Wf = (const float*)d_in[6];
    const float* bf = (const float*)d_in[7];
    float* out = (float*)d_out;

    const int N = in_sizes[0] / D_IN;        // 100000
    const int E = in_sizes[1] / 2;           // 3200000
    const int* src = edge;
    const int* dst = edge + E;

    float* ws    = (float*)d_ws;
    float* dinv  = ws;                                   // N
    float* bufH  = dinv + N;                             // N*32
    float* bufG  = bufH + (size_t)N * D_H;               // N*32
    float* bufF  = bufG + (size_t)N * D_H;               // N*40
    float* stats = bufF + (size_t)N * N_CLS;             // 33+

    const int TB = 256;
    dim3 blk(TB);
    int gN    = (N + TB - 1) / TB;
    int gE    = (E + TB - 1) / TB;
    int gNC   = ((size_t)N * D_H + TB - 1) / TB;
    int gE32  = ((size_t)E * 32 + TB - 1) / TB;
    int gGemm = (N + 127) / 128;                         // 8 waves x 16 rows per block
    int gNF   = ((size_t)N * N_CLS + TB - 1) / TB;

    // degrees / dinv
    k_fill1<<<gN, blk, 0, stream>>>(dinv, N);
    k_deg<<<gE, blk, 0, stream>>>(dst, dinv, E);
    k_rsqrt<<<gN, blk, 0, stream>>>(dinv, N);

    // ---- layer 0 ----
    k_gemm_wmma<D_IN><<<gGemm, blk, 0, stream>>>(x, W0, bufH, N);
    k_initagg<D_H><<<gNC, blk, 0, stream>>>(bufH, dinv, b0, bufG, N);
    k_scatter<D_H><<<gE32, blk, 0, stream>>>(bufH, src, dst, dinv, bufG, E);
    hipMemsetAsync(stats, 0, 64 * sizeof(float), stream);
    k_stats<<<512, blk, 0, stream>>>(bufG, stats, N);
    k_finalize<<<1, 1, 0, stream>>>(stats, N);
    k_applypn<<<gNC, blk, 0, stream>>>(bufG, stats, N);

    // ---- layer 1 ----
    k_gemm_wmma<D_H><<<gGemm, blk, 0, stream>>>(bufG, W1, bufH, N);
    k_initagg<D_H><<<gNC, blk, 0, stream>>>(bufH, dinv, b1, bufG, N);
    k_scatter<D_H><<<gE32, blk, 0, stream>>>(bufH, src, dst, dinv, bufG, E);
    hipMemsetAsync(stats, 0, 64 * sizeof(float), stream);
    k_stats<<<512, blk, 0, stream>>>(bufG, stats, N);
    k_finalize<<<1, 1, 0, stream>>>(stats, N);
    k_applypn<<<gNC, blk, 0, stream>>>(bufG, stats, N);

    // ---- final layer (no pairnorm/relu) ----
    k_gemm_final<<<gNF, blk, 0, stream>>>(bufG, Wf, bufF, N);
    k_initagg<N_CLS><<<gNF, blk, 0, stream>>>(bufF, dinv, bf, out, N);
    k_scatter<N_CLS><<<gE32, blk, 0, stream>>>(bufF, src, dst, dinv, out, E);
}